// MambaLM_62448824483845
// MI455X (gfx1250) — compile-verified
//
#include <hip/hip_runtime.h>
#include <hip/hip_bf16.h>

// ---------------- problem constants (match reference) ----------------
#define V_    32000
#define DM    1024
#define NL_   4
#define NST   16
#define KC    4
#define DI    2048
#define T_    2048
#define EPS_  1e-5f
#define XPD   33        // 2N+1

typedef __attribute__((ext_vector_type(2))) float v2f;
typedef __attribute__((ext_vector_type(8))) float v8f;

// ---------------- embedding gather ----------------
__global__ void embed_kernel(const int* __restrict__ idx,
                             const float* __restrict__ emb,
                             float* __restrict__ x) {
  int t = blockIdx.x;
  int row = idx[t];
  const float4* src = (const float4*)(emb + (size_t)row * DM);
  float4* dst = (float4*)(x + (size_t)t * DM);
  dst[threadIdx.x] = src[threadIdx.x];   // 256 threads * 4 floats = 1024 = DM
}

// ---------------- RMSNorm (one block per row) ----------------
__global__ void rmsnorm_kernel(const float* __restrict__ x,
                               const float* __restrict__ w,
                               float* __restrict__ out) {
  int t = blockIdx.x;
  const float* xr = x + (size_t)t * DM;
  float v[4];
  float s = 0.f;
#pragma unroll
  for (int i = 0; i < 4; i++) {
    v[i] = xr[threadIdx.x + i * 256];
    s += v[i] * v[i];
  }
#pragma unroll
  for (int m = 16; m >= 1; m >>= 1) s += __shfl_xor(s, m, 32);
  __shared__ float red[8];
  int wid = threadIdx.x >> 5, lane = threadIdx.x & 31;
  if (lane == 0) red[wid] = s;
  __syncthreads();
  float tot = 0.f;
#pragma unroll
  for (int i = 0; i < 8; i++) tot += red[i];
  float rinv = rsqrtf(tot / (float)DM + EPS_);
#pragma unroll
  for (int i = 0; i < 4; i++) {
    int c = threadIdx.x + i * 256;
    out[(size_t)t * DM + c] = v[i] * rinv * w[c];
  }
}

// ---------------- WMMA fp32 GEMM: C[M x Nd] = A[M x Kd] * W[Nd x Kd]^T (+resid) ----------------
// block = 256 threads = 8 waves (wave32). Block tile 128(M) x 128(N).
// Wave grid 4(M) x 2(N); each wave computes 32x64 = 2x4 WMMA 16x16 tiles.
// Double-buffered LDS; tiles staged via CDNA5 async global->LDS DMA.
// Next tile's DMA is issued before computing the current one, then
// s_wait_asynccnt 0x4 (per-wave in-order completion => current tile ready),
// so the global->LDS latency hides behind 32 WMMAs per chunk per wave.
#define TM 128
#define TN 128
#define TK 16
#define LDP (TK + 4)
#define ABUF (TM * LDP)   // floats per A buffer
#define BBUF (TN * LDP)   // floats per B buffer

__global__ __launch_bounds__(256) void gemm_nt_wmma(
    const float* __restrict__ A, const float* __restrict__ W,
    const float* resid, float* C, int M, int Nd, int Kd)
{
  __shared__ __align__(16) float As[2 * ABUF];
  __shared__ __align__(16) float Bs[2 * BBUF];

  const int tid  = threadIdx.x;
  const int wid  = tid >> 5;
  const int lane = tid & 31;
  const int half = lane >> 4;       // 0: lanes 0-15, 1: lanes 16-31
  const int lr   = lane & 15;
  const int wm   = wid & 3;         // wave row (M): 0..3
  const int wn   = wid >> 2;        // wave col (N): 0..1
  const int m0   = blockIdx.y * TM;
  const int n0   = blockIdx.x * TN;

  v8f acc[2][4] = {};               // zero-init accumulators (32x64 per wave)

  const int srow = tid >> 2;        // 0..63
  const int cg   = tid & 3;         // float4 column group in K-chunk

  // per-thread LDS byte offsets for the async DMA destinations (buffer 0)
  // (generic LDS pointers carry the LDS byte offset in their low 32 bits)
  const unsigned ldsA0 = (unsigned)(size_t)&As[(srow     ) * LDP + cg * 4];
  const unsigned ldsA1 = (unsigned)(size_t)&As[(srow + 64) * LDP + cg * 4];
  const unsigned ldsB0 = (unsigned)(size_t)&Bs[(srow     ) * LDP + cg * 4];
  const unsigned ldsB1 = (unsigned)(size_t)&Bs[(srow + 64) * LDP + cg * 4];

  const float* gA0 = A + (size_t)(m0 + srow     ) * Kd + cg * 4;
  const float* gA1 = A + (size_t)(m0 + srow + 64) * Kd + cg * 4;
  const float* gB0 = W + (size_t)(n0 + srow     ) * Kd + cg * 4;
  const float* gB1 = W + (size_t)(n0 + srow + 64) * Kd + cg * 4;

  auto issue_tile = [&](int buf, int k) {
    const unsigned aoff = (unsigned)(buf * ABUF * 4);
    const unsigned boff = (unsigned)(buf * BBUF * 4);
    asm volatile("global_load_async_to_lds_b128 %0, %1, off"
                 :: "v"(ldsA0 + aoff), "v"((unsigned long long)(size_t)(gA0 + k)) : "memory");
    asm volatile("global_load_async_to_lds_b128 %0, %1, off"
                 :: "v"(ldsA1 + aoff), "v"((unsigned long long)(size_t)(gA1 + k)) : "memory");
    asm volatile("global_load_async_to_lds_b128 %0, %1, off"
                 :: "v"(ldsB0 + boff), "v"((unsigned long long)(size_t)(gB0 + k)) : "memory");
    asm volatile("global_load_async_to_lds_b128 %0, %1, off"
                 :: "v"(ldsB1 + boff), "v"((unsigned long long)(size_t)(gB1 + k)) : "memory");
  };

  issue_tile(0, 0);                 // prologue: tile 0 into buffer 0

  int sel = 0;
  for (int k0 = 0; k0 < Kd; k0 += TK) {
    // ---- prefetch next tile into the alternate buffer, then wait for current ----
    if (k0 + TK < Kd) {
      issue_tile(sel ^ 1, k0 + TK);
      asm volatile("s_wait_asynccnt 0x4" ::: "memory");  // current tile's 4 loads done
    } else {
      asm volatile("s_wait_asynccnt 0x0" ::: "memory");
    }
    __syncthreads();

    const float* Asb = As + sel * ABUF;
    const float* Bsb = Bs + sel * BBUF;

    // ---- 4 k-steps of 4 per chunk; 32 WMMAs per chunk per wave ----
#pragma unroll
    for (int kk = 0; kk < TK; kk += 4) {
      const int kc = kk + 2 * half;  // ISA layout: lanes16-31 hold K+2..K+3
      v2f a0 = *(const v2f*)(&Asb[(wm * 32 +      lr) * LDP + kc]);
      v2f a1 = *(const v2f*)(&Asb[(wm * 32 + 16 + lr) * LDP + kc]);
      v2f b0 = *(const v2f*)(&Bsb[(wn * 64 +      lr) * LDP + kc]);
      v2f b1 = *(const v2f*)(&Bsb[(wn * 64 + 16 + lr) * LDP + kc]);
      v2f b2 = *(const v2f*)(&Bsb[(wn * 64 + 32 + lr) * LDP + kc]);
      v2f b3 = *(const v2f*)(&Bsb[(wn * 64 + 48 + lr) * LDP + kc]);
      acc[0][0] = __builtin_amdgcn_wmma_f32_16x16x4_f32(false, a0, false, b0, (short)0, acc[0][0], false, false);
      acc[0][1] = __builtin_amdgcn_wmma_f32_16x16x4_f32(false, a0, false, b1, (short)0, acc[0][1], false, false);
      acc[0][2] = __builtin_amdgcn_wmma_f32_16x16x4_f32(false, a0, false, b2, (short)0, acc[0][2], false, false);
      acc[0][3] = __builtin_amdgcn_wmma_f32_16x16x4_f32(false, a0, false, b3, (short)0, acc[0][3], false, false);
      acc[1][0] = __builtin_amdgcn_wmma_f32_16x16x4_f32(false, a1, false, b0, (short)0, acc[1][0], false, false);
      acc[1][1] = __builtin_amdgcn_wmma_f32_16x16x4_f32(false, a1, false, b1, (short)0, acc[1][1], false, false);
      acc[1][2] = __builtin_amdgcn_wmma_f32_16x16x4_f32(false, a1, false, b2, (short)0, acc[1][2], false, false);
      acc[1][3] = __builtin_amdgcn_wmma_f32_16x16x4_f32(false, a1, false, b3, (short)0, acc[1][3], false, false);
    }
    __syncthreads();                // release this buffer for the DMA after next
    sel ^= 1;
  }

  // ---- writeback (C/D layout: VGPR r -> M = r + half*8, N = lr) ----
#pragma unroll
  for (int mt = 0; mt < 2; mt++)
#pragma unroll
    for (int nt = 0; nt < 4; nt++)
#pragma unroll
      for (int r = 0; r < 8; r++) {
        int row = m0 + wm * 32 + mt * 16 + half * 8 + r;
        int col = n0 + wn * 64 + nt * 16 + lr;
        size_t off = (size_t)row * Nd + col;
        float vacc = acc[mt][nt][r];
        if (resid) vacc += resid[off];
        C[off] = vacc;
      }
}

// ---------------- depthwise causal conv (K=4) + bias + SiLU ----------------
__global__ void conv_silu_kernel(const float* __restrict__ xz,
                                 const float* __restrict__ cw,
                                 const float* __restrict__ cb,
                                 float* __restrict__ u) {
  int g = blockIdx.x * 256 + threadIdx.x;   // t*DI + d
  int t = g >> 11;                          // / DI
  int d = g & (DI - 1);
  float acc = cb[d];
#pragma unroll
  for (int j = 0; j < KC; j++) {
    int tt = t - (KC - 1) + j;
    if (tt >= 0) acc += cw[d * KC + j] * xz[(size_t)tt * (2 * DI) + d];
  }
  u[g] = acc / (1.f + expf(-acc));          // SiLU
}

// ---------------- x_proj: xp[T x 33] = u[T x DI] @ xp_w[33 x DI]^T ----------------
__global__ void xproj_kernel(const float* __restrict__ u,
                             const float* __restrict__ xw,
                             float* __restrict__ xp) {
  int t = blockIdx.x;
  int wid = threadIdx.x >> 5, lane = threadIdx.x & 31;
  const float* ur = u + (size_t)t * DI;
  for (int n = wid; n < XPD; n += 8) {
    const float* wr = xw + (size_t)n * DI;
    float s = 0.f;
    for (int k = lane; k < DI; k += 32) s += ur[k] * wr[k];
#pragma unroll
    for (int m = 16; m >= 1; m >>= 1) s += __shfl_xor(s, m, 32);
    if (lane == 0) xp[(size_t)t * XPD + n] = s;
  }
}

// ---------------- selective scan + softplus(dt) + D skip + SiLU gate, fused ----------------
// one thread per channel d; 16-entry state in registers; sequential over T.
__global__ void scan_gate_kernel(const float* __restrict__ u,
                                 const float* __restrict__ xp,
                                 const float* __restrict__ xz,
                                 const float* __restrict__ A_log,
                                 const float* __restrict__ dt_w,
                                 const float* __restrict__ dt_b,
                                 const float* __restrict__ Dp,
                                 float* __restrict__ y) {
  int d = blockIdx.x * 256 + threadIdx.x;   // 0..DI-1
  float A[NST];
#pragma unroll
  for (int n = 0; n < NST; n++) A[n] = -expf(A_log[(size_t)d * NST + n]);
  float dtw = dt_w[d], dtb = dt_b[d], Dd = Dp[d];
  float h[NST] = {};
  for (int t = 0; t < T_; t++) {
    const float* xpr = xp + (size_t)t * XPD;
    float dtv = xpr[0] * dtw + dtb;
    dtv = (dtv > 20.f) ? dtv : log1pf(expf(dtv));   // softplus
    float ut = u[(size_t)t * DI + d];
    float du = dtv * ut;
    float acc = 0.f;
#pragma unroll
    for (int n = 0; n < NST; n++) {
      float dA = expf(dtv * A[n]);
      h[n] = dA * h[n] + du * xpr[1 + n];           // Bv
      acc += h[n] * xpr[1 + NST + n];               // Cv
    }
    float zv = xz[(size_t)t * (2 * DI) + DI + d];
    float yv = (acc + ut * Dd) * (zv / (1.f + expf(-zv)));
    y[(size_t)t * DI + d] = yv;
  }
}

// ---------------- host-side pipeline ----------------
extern "C" void kernel_launch(void* const* d_in, const int* in_sizes, int n_in,
                              void* d_out, int out_size, void* d_ws, size_t ws_size,
                              hipStream_t stream) {
  const int*   idx   = (const int*)  d_in[0];
  const float* emb   = (const float*)d_in[1];
  const float* normw = (const float*)d_in[2];
  const float* inw   = (const float*)d_in[3];
  const float* convw = (const float*)d_in[4];
  const float* convb = (const float*)d_in[5];
  const float* xpw   = (const float*)d_in[6];
  const float* dtw   = (const float*)d_in[7];
  const float* dtb   = (const float*)d_in[8];
  const float* Alog  = (const float*)d_in[9];
  const float* Dpar  = (const float*)d_in[10];
  const float* outw  = (const float*)d_in[11];
  const float* normf = (const float*)d_in[12];
  float* out = (float*)d_out;

  float* ws = (float*)d_ws;
  float* x  = ws;                               // T x DM
  float* xn = x  + (size_t)T_ * DM;             // T x DM
  float* xz = xn + (size_t)T_ * DM;             // T x 2*DI
  float* u  = xz + (size_t)T_ * 2 * DI;         // T x DI
  float* xp = u  + (size_t)T_ * DI;             // T x 33
  float* y  = xp + (size_t)T_ * XPD;            // T x DI

  embed_kernel<<<T_, 256, 0, stream>>>(idx, emb, x);

  for (int l = 0; l < NL_; l++) {
    rmsnorm_kernel<<<T_, 256, 0, stream>>>(x, normw + (size_t)l * DM, xn);
    // in_proj: xz = xn @ in_w^T  (2048 x 1024 x 4096)
    gemm_nt_wmma<<<dim3((2 * DI) / TN, T_ / TM), 256, 0, stream>>>(
        xn, inw + (size_t)l * 2 * DI * DM, nullptr, xz, T_, 2 * DI, DM);
    conv_silu_kernel<<<(T_ * DI) / 256, 256, 0, stream>>>(
        xz, convw + (size_t)l * DI * KC, convb + (size_t)l * DI, u);
    xproj_kernel<<<T_, 256, 0, stream>>>(u, xpw + (size_t)l * XPD * DI, xp);
    scan_gate_kernel<<<DI / 256, 256, 0, stream>>>(
        u, xp, xz, Alog + (size_t)l * DI * NST,
        dtw + (size_t)l * DI, dtb + (size_t)l * DI, Dpar + (size_t)l * DI, y);
    // out_proj + residual: x = x + y @ out_w^T  (2048 x 2048 x 1024)
    gemm_nt_wmma<<<dim3(DM / TN, T_ / TM), 256, 0, stream>>>(
        y, outw + (size_t)l * DM * DI, x, x, T_, DM, DI);
  }

  rmsnorm_kernel<<<T_, 256, 0, stream>>>(x, normf, xn);
  // logits: out = xn @ emb^T  (2048 x 1024 x 32000), 250 x 16 blocks
  gemm_nt_wmma<<<dim3(V_ / TN, T_ / TM), 256, 0, stream>>>(
      xn, emb, nullptr, out, T_, V_, DM);
}